// LCM_50199577756243
// MI455X (gfx1250) — compile-verified
//
#include <hip/hip_runtime.h>
#include <hip/hip_bf16.h>
#include <math.h>

#define N_DATASETS 16
#define POLY_DEG   3
#define N_AGES     101
#define K_DIM      202                        // 2 * N_AGES
#define DK         (N_DATASETS * K_DIM)       // 3232
#define K_PAD      208                        // row stride in LDS, 832B = 16-aligned
#define DK_PAD     (N_DATASETS * K_PAD)       // 3328 = 13 * 256 exactly
#define EPSV       1e-4f

#define WAVES_PER_BLOCK 8
#define THREADS   (WAVES_PER_BLOCK * 32)      // 256
#define GRID_MAIN 2048

typedef float v4f __attribute__((ext_vector_type(4)));
typedef float v2f __attribute__((ext_vector_type(2)));

// ---------------------------------------------------------------------------
// Kernel 1: per-dataset rowsum table [16, 202] -> d_ws. Tiny, accuracy-first.
// ---------------------------------------------------------------------------
__global__ void rowsum_table_kernel(const float* __restrict__ zs,
                                    const float* __restrict__ us,
                                    const float* __restrict__ vs,
                                    const float* __restrict__ poly,
                                    const float* __restrict__ dages,
                                    const float* __restrict__ a_hat,
                                    const float* __restrict__ g_hat,
                                    float* __restrict__ out) {
    int idx = blockIdx.x * blockDim.x + threadIdx.x;
    if (idx >= DK) return;
    int d = idx / K_DIM;
    int k = idx - d * K_DIM;
    int c = k / N_AGES;
    int a = k - c * N_AGES;
    int zb = (d * 2 + c) * POLY_DEG;

    float p0 = poly[0 * N_AGES + a];
    float p1 = poly[1 * N_AGES + a];
    float p2 = poly[2 * N_AGES + a];

    float zeta = zs[zb] * p0 + zs[zb + 1] * p1 + zs[zb + 2] * p2;
    float mu   = us[zb] * p0 + us[zb + 1] * p1 + us[zb + 2] * p2;
    float sg   = vs[zb] * p0 + vs[zb + 1] * p1 + vs[zb + 2] * p2;
    float var  = sg * sg + EPSV;
    float c2   = -0.5f / var;

    float norm = 0.0f;
    for (int m = 0; m < N_AGES; ++m) {
        float t = mu - dages[d * N_AGES + m];
        norm += expf(t * t * c2);
    }
    norm += EPSV;

    float num = 0.0f;
    for (int a2 = 0; a2 < N_AGES; ++a2) {
        float t = mu - a_hat[a2];
        num += expf(t * t * c2);
    }

    float g0 = g_hat[0], g1 = g_hat[1];
    float pg = 1.0f / (1.0f + expf(-g0 * zeta)) + 1.0f / (1.0f + expf(-g1 * zeta));

    out[idx] = (num / norm) * pg;
}

// ---------------------------------------------------------------------------
// Kernel 2: streaming softmax * gathered table row. One row per wave32.
// Row layout per lane: b128 (cols 4l..4l+3) + b64 (cols 128+2l..129+2l)
//                      + masked b32 (col 192+l, lanes 0..9)  => 7 values/lane.
// Table preloaded to LDS (padded stride 208) via CDNA5 async global->LDS DMA.
// ---------------------------------------------------------------------------
__global__ __launch_bounds__(THREADS) void softmax_scale_kernel(
        const float* __restrict__ input,
        const int*   __restrict__ dsid,
        const float* __restrict__ rowsum,   // [DK] table in d_ws (global)
        float*       __restrict__ out,
        int n_rows) {
    __shared__ __align__(16) float table[DK_PAD];

    // Async copy the table into LDS (tracked by ASYNCcnt). 13 uniform iters.
    for (int i = threadIdx.x; i < DK_PAD; i += THREADS) {
        int d = i / K_PAD;
        int k = i - d * K_PAD;
        if (k < K_DIM) {
            unsigned     loff = (unsigned)(unsigned long long)(table + i);
            const float* gsrc = rowsum + d * K_DIM + k;
            asm volatile("global_load_async_to_lds_b32 %0, %1, off"
                         :: "v"(loff), "v"(gsrc) : "memory");
        }
    }
    asm volatile("s_wait_asynccnt 0" ::: "memory");
    __syncthreads();

    const int lane    = threadIdx.x & 31;
    const int wave    = threadIdx.x >> 5;
    const int wstride = gridDim.x * WAVES_PER_BLOCK;
    const bool tailln = (lane < 10);
    const float NEGINF = -__builtin_huge_valf();

    for (int row = blockIdx.x * WAVES_PER_BLOCK + wave; row < n_rows; row += wstride) {
        const float* rp = input + (size_t)row * K_DIM;

        // Single-pass vector load of the row (no reuse -> non-temporal).
        v4f xa = __builtin_nontemporal_load((const v4f*)(rp + 4 * lane));
        v2f xb = __builtin_nontemporal_load((const v2f*)(rp + 128 + 2 * lane));
        float xc = tailln ? __builtin_nontemporal_load(rp + 192 + lane) : NEGINF;

        // Row max: local then wave32 xor-shuffle reduction.
        float m = fmaxf(fmaxf(fmaxf(xa.x, xa.y), fmaxf(xa.z, xa.w)),
                        fmaxf(fmaxf(xb.x, xb.y), xc));
#pragma unroll
        for (int s = 16; s > 0; s >>= 1) m = fmaxf(m, __shfl_xor(m, s, 32));

        // exp(x - max) and row sum.
        xa.x = __expf(xa.x - m);  xa.y = __expf(xa.y - m);
        xa.z = __expf(xa.z - m);  xa.w = __expf(xa.w - m);
        xb.x = __expf(xb.x - m);  xb.y = __expf(xb.y - m);
        xc   = tailln ? __expf(xc - m) : 0.0f;
        float sum = ((xa.x + xa.y) + (xa.z + xa.w)) + ((xb.x + xb.y) + xc);
#pragma unroll
        for (int s = 16; s > 0; s >>= 1) sum += __shfl_xor(sum, s, 32);
        float inv = 1.0f / sum;

        // Scale by LDS-resident table row for this sample's dataset.
        int d = dsid[row];
        const float* trow = table + d * K_PAD;             // 16B-aligned base
        v4f  ta = *(const v4f*)(trow + 4 * lane);          // ds_load_b128
        v2f  tb = *(const v2f*)(trow + 128 + 2 * lane);    // ds_load_b64
        float tc = tailln ? trow[192 + lane] : 0.0f;

        float* op = out + (size_t)row * K_DIM;
        v4f ra; v2f rb;
        ra.x = xa.x * inv * ta.x;  ra.y = xa.y * inv * ta.y;
        ra.z = xa.z * inv * ta.z;  ra.w = xa.w * inv * ta.w;
        rb.x = xb.x * inv * tb.x;  rb.y = xb.y * inv * tb.y;
        __builtin_nontemporal_store(ra, (v4f*)(op + 4 * lane));
        __builtin_nontemporal_store(rb, (v2f*)(op + 128 + 2 * lane));
        if (tailln) {
            __builtin_nontemporal_store(xc * inv * tc, op + 192 + lane);
        }
    }
}

// ---------------------------------------------------------------------------
extern "C" void kernel_launch(void* const* d_in, const int* in_sizes, int n_in,
                              void* d_out, int out_size, void* d_ws, size_t ws_size,
                              hipStream_t stream) {
    const float* input = (const float*)d_in[0];
    const int*   dsid  = (const int*)  d_in[1];
    const float* zs    = (const float*)d_in[2];
    const float* us    = (const float*)d_in[3];
    const float* vs    = (const float*)d_in[4];
    const float* poly  = (const float*)d_in[5];
    const float* dages = (const float*)d_in[6];
    const float* a_hat = (const float*)d_in[7];
    const float* g_hat = (const float*)d_in[8];

    float* outp   = (float*)d_out;
    float* rowsum = (float*)d_ws;                 // DK floats of scratch
    int n_rows = in_sizes[1];                     // B = 262144

    rowsum_table_kernel<<<(DK + 255) / 256, 256, 0, stream>>>(
        zs, us, vs, poly, dages, a_hat, g_hat, rowsum);

    softmax_scale_kernel<<<GRID_MAIN, THREADS, 0, stream>>>(
        input, dsid, rowsum, outp, n_rows);
}